// BasinCoupledQFIAttention_30021821399807
// MI455X (gfx1250) — compile-verified
//
#include <hip/hip_runtime.h>
#include <math.h>

// Problem constants (B, T, D fixed by the reference).
#define BB   4
#define TT   512
#define DD   128
#define KAUG 256            // augmented K: [sqrt(pn) | sqrt(EPS/2)*rsqrt(pn)]
#define EPSF 1e-8f

typedef float v2f __attribute__((ext_vector_type(2)));
typedef float v8f __attribute__((ext_vector_type(8)));

// ---------------------------------------------------------------------------
// Kernel 1: scalar inverse-temperature = 1 / max(sigmoid(basin.w + b)+0.5, 1e-6)
// ---------------------------------------------------------------------------
__global__ __launch_bounds__(32)
void qfi_temp_kernel(const float* __restrict__ basin,
                     const float* __restrict__ w_temp,
                     const float* __restrict__ b_temp,
                     float* __restrict__ ws_scalar) {
    int lane = threadIdx.x;
    float s = 0.0f;
    for (int d = lane; d < DD; d += 32) s += basin[d] * w_temp[d];
    #pragma unroll
    for (int off = 16; off; off >>= 1) s += __shfl_xor(s, off, 32);
    if (lane == 0) {
        float z = s + b_temp[0];
        float t = 1.0f / (1.0f + expf(-z)) + 0.5f;
        t = fmaxf(t, 1e-6f);
        ws_scalar[0] = 1.0f / t;   // store inverse temperature
    }
}

// ---------------------------------------------------------------------------
// Kernel 2: per-row simplex projection -> augmented sqrt matrix Qa (B*T, 256)
//   Qa[:, 0:128]   = sqrt(pn)
//   Qa[:, 128:256] = sqrt(EPS/2) * rsqrt(pn)
// so that inner = Qa Qa^T = sum sqrt(pi*pj) + (EPS/2) sum 1/sqrt(pi*pj)
//               ~= sum sqrt(pi*pj + EPS)   (rel. error ~1e-11, below fp32 ulp)
// ---------------------------------------------------------------------------
__global__ __launch_bounds__(128)
void qfi_prep_kernel(const float* __restrict__ x, float* __restrict__ Qa) {
    const int row = blockIdx.x;      // b*T + t
    const int d   = threadIdx.x;     // 0..127
    const int wid = threadIdx.x >> 5;
    const int ln  = threadIdx.x & 31;
    __shared__ float red[4];

    float v = x[row * DD + d];
    // numerically stable softplus: max(x,0) + log1p(exp(-|x|))
    float p = fmaxf(v, 0.0f) + log1pf(expf(-fabsf(v)));

    float s = p;
    #pragma unroll
    for (int off = 16; off; off >>= 1) s += __shfl_xor(s, off, 32);
    if (ln == 0) red[wid] = s;
    __syncthreads();
    float s1 = red[0] + red[1] + red[2] + red[3];
    __syncthreads();

    float p1 = fmaxf(p / (s1 + EPSF), EPSF);

    s = p1;
    #pragma unroll
    for (int off = 16; off; off >>= 1) s += __shfl_xor(s, off, 32);
    if (ln == 0) red[wid] = s;
    __syncthreads();
    float s2 = red[0] + red[1] + red[2] + red[3];

    float pn = p1 / (s2 + EPSF);
    Qa[(size_t)row * KAUG + d]      = sqrtf(pn);
    Qa[(size_t)row * KAUG + DD + d] = sqrtf(0.5f * EPSF / pn);
}

// ---------------------------------------------------------------------------
// Kernel 3: fused  inner = Qa Qa^T (f32 WMMA, K=256)
//                  -> d = 2 acos(clip(inner)); logits = -d * invTemp
//                  -> row softmax (shfl + LDS reductions)
//                  -> x_attn = w @ x (f32 WMMA, K=512)
//                  -> out = x + rs * (x_attn - x)
// One workgroup (4 waves) per 16 output rows. Wave w covers j-cols [128w,128w+128)
// in GEMM1 and output cols [32w, 32w+32) in GEMM2.
// ---------------------------------------------------------------------------
__global__ __launch_bounds__(128)
void qfi_attn_kernel(const float* __restrict__ x,
                     const float* __restrict__ Qa,
                     const float* __restrict__ invTempPtr,
                     const float* __restrict__ rsPtr,
                     float* __restrict__ out) {
    const int b    = blockIdx.x >> 5;          // TT/16 == 32 row-blocks per batch
    const int i0   = (blockIdx.x & 31) << 4;   // first of 16 rows
    const int tid  = threadIdx.x;
    const int wave = tid >> 5;
    const int lane = tid & 31;
    const int lm   = lane & 15;                // col-in-tile / row-in-A
    const int lh   = lane >> 4;                // half selector

    __shared__ float lds_w[16][516];           // softmax weights, padded stride
    __shared__ float redbuf[4][16];            // cross-wave row reductions

    const float invTemp = invTempPtr[0];
    const float rs      = rsPtr[0];

    // ---------------- GEMM1: inner = Qa Qa^T over K=256 ----------------
    const float* Qb   = Qa + (size_t)b * TT * KAUG;
    const float* Arow = Qb + (size_t)(i0 + lm) * KAUG + lh * 2;   // A: row=lm, k=lh*2+{0,1}
    const int jBase = wave * 128;
    const float* Brow[8];
    #pragma unroll
    for (int t = 0; t < 8; ++t)
        Brow[t] = Qb + (size_t)(jBase + t * 16 + lm) * KAUG + lh * 2;

    v8f acc[8] = {};
    for (int k0 = 0; k0 < KAUG; k0 += 4) {
        v2f a = *(const v2f*)(Arow + k0);
        #pragma unroll
        for (int t = 0; t < 8; ++t) {
            v2f bf = *(const v2f*)(Brow[t] + k0);
            acc[t] = __builtin_amdgcn_wmma_f32_16x16x4_f32(
                false, a, false, bf, (short)0, acc[t], false, false);
        }
    }

    // ---------------- logits + row max ----------------
    float rowmax[8], rowsum[8];
    #pragma unroll
    for (int e = 0; e < 8; ++e) {
        float m = -3.4e38f;
        #pragma unroll
        for (int t = 0; t < 8; ++t) {
            float v = acc[t][e];
            v = fminf(fmaxf(v, -1.0f + 1e-6f), 1.0f - 1e-6f);
            v = -2.0f * acosf(v) * invTemp;
            acc[t][e] = v;
            m = fmaxf(m, v);
        }
        // reduce across the 16 lanes of each half (rows stay separated)
        #pragma unroll
        for (int off = 8; off; off >>= 1) m = fmaxf(m, __shfl_xor(m, off, 32));
        rowmax[e] = m;
    }
    if (lm == 0) {
        #pragma unroll
        for (int e = 0; e < 8; ++e) redbuf[wave][e + lh * 8] = rowmax[e];
    }
    __syncthreads();
    #pragma unroll
    for (int e = 0; e < 8; ++e) {
        int r = e + lh * 8;
        rowmax[e] = fmaxf(fmaxf(redbuf[0][r], redbuf[1][r]),
                          fmaxf(redbuf[2][r], redbuf[3][r]));
    }
    __syncthreads();

    // ---------------- exp + row sum ----------------
    #pragma unroll
    for (int e = 0; e < 8; ++e) {
        float s = 0.0f;
        #pragma unroll
        for (int t = 0; t < 8; ++t) {
            float ev = expf(acc[t][e] - rowmax[e]);
            acc[t][e] = ev;
            s += ev;
        }
        #pragma unroll
        for (int off = 8; off; off >>= 1) s += __shfl_xor(s, off, 32);
        rowsum[e] = s;
    }
    if (lm == 0) {
        #pragma unroll
        for (int e = 0; e < 8; ++e) redbuf[wave][e + lh * 8] = rowsum[e];
    }
    __syncthreads();
    #pragma unroll
    for (int e = 0; e < 8; ++e) {
        int r = e + lh * 8;
        rowsum[e] = redbuf[0][r] + redbuf[1][r] + redbuf[2][r] + redbuf[3][r];
    }

    // normalized weights -> LDS (row-major, padded)
    #pragma unroll
    for (int e = 0; e < 8; ++e) {
        float inv = 1.0f / rowsum[e];
        int r = e + lh * 8;
        #pragma unroll
        for (int t = 0; t < 8; ++t)
            lds_w[r][jBase + t * 16 + lm] = acc[t][e] * inv;
    }
    __syncthreads();

    // ---------------- GEMM2: x_attn = w @ x over K=512 ----------------
    const int n0 = wave * 32;
    const float* Xb = x + (size_t)b * TT * DD;
    const float* wrow = &lds_w[lm][lh * 2];    // A: row=lm, k=lh*2+{0,1}
    v8f acc2[2] = {};
    for (int k0 = 0; k0 < TT; k0 += 4) {
        v2f a = *(const v2f*)(wrow + k0);
        #pragma unroll
        for (int t = 0; t < 2; ++t) {
            int n = n0 + t * 16 + lm;
            v2f bf;
            bf.x = Xb[(size_t)(k0 + lh * 2)     * DD + n];
            bf.y = Xb[(size_t)(k0 + lh * 2 + 1) * DD + n];
            acc2[t] = __builtin_amdgcn_wmma_f32_16x16x4_f32(
                false, a, false, bf, (short)0, acc2[t], false, false);
        }
    }

    // ---------------- residual epilogue ----------------
    #pragma unroll
    for (int t = 0; t < 2; ++t) {
        #pragma unroll
        for (int e = 0; e < 8; ++e) {
            int row = i0 + e + lh * 8;
            int col = n0 + t * 16 + lm;
            size_t idx = (size_t)b * TT * DD + (size_t)row * DD + col;
            float xv = x[idx];
            out[idx] = xv + rs * (acc2[t][e] - xv);
        }
    }
}

// ---------------------------------------------------------------------------
extern "C" void kernel_launch(void* const* d_in, const int* in_sizes, int n_in,
                              void* d_out, int out_size, void* d_ws, size_t ws_size,
                              hipStream_t stream) {
    (void)in_sizes; (void)n_in; (void)out_size; (void)ws_size;
    const float* x      = (const float*)d_in[0];
    const float* basin  = (const float*)d_in[1];
    const float* w_temp = (const float*)d_in[2];
    const float* b_temp = (const float*)d_in[3];
    const float* rscale = (const float*)d_in[4];
    float* out = (float*)d_out;

    float* wsf = (float*)d_ws;
    float* scalarBuf = wsf;            // [0] = inverse temperature
    float* Qa        = wsf + 16;       // 64B-aligned; B*T*KAUG floats = 2 MB

    qfi_temp_kernel<<<1, 32, 0, stream>>>(basin, w_temp, b_temp, scalarBuf);
    qfi_prep_kernel<<<BB * TT, 128, 0, stream>>>(x, Qa);
    qfi_attn_kernel<<<BB * (TT / 16), 128, 0, stream>>>(x, Qa, scalarBuf, rscale, out);
}